// EdgeConsistencyLoss_57827439674028
// MI455X (gfx1250) — compile-verified
//
#include <hip/hip_runtime.h>
#include <math.h>

// ---------------------------------------------------------------------------
// EdgeConsistencyLoss, fused single pass + deterministic reduction.
//   inputs/targets: [B,1,1024,1024] f32.  Output: scalar f32 mean(diff^2).
// Memory-bound: 128 MB in, 4 B out -> one HBM pass is optimal. Tile in LDS,
// compute everything (sobel, edges, maxpool, diff^2) in-registers/LDS.
// Wave-level sum is done on the matrix pipe with V_WMMA_F32_16X16X4_F32
// against a ones B-matrix (exact f32 reduction).
// ---------------------------------------------------------------------------

typedef float v2f __attribute__((ext_vector_type(2)));
typedef float v8f __attribute__((ext_vector_type(8)));

#define IMG_H 1024
#define IMG_W 1024
#define TILE  32

__global__ __launch_bounds__(256)
void edge_loss_tile(const float* __restrict__ inp,
                    const float* __restrict__ tgt,
                    float* __restrict__ partial)
{
    // targets need halo 2 (edge-of-edge for maxpool), inputs halo 1.
    __shared__ float st[36][37];   // targets tile, halo 2  (+1 col pad vs banks)
    __shared__ float si[34][35];   // inputs  tile, halo 1
    __shared__ float te[34][35];   // target_edge on halo-1 region
    __shared__ float wsum[8];

    const int tid = threadIdx.x;
    const int blk = blockIdx.x;
    const int b   = blk >> 10;            // 1024 tiles per image (32x32 grid)
    const int t   = blk & 1023;
    const int y0  = (t >> 5) << 5;
    const int x0  = (t & 31) << 5;

    const float* timg = tgt + (size_t)b * (IMG_H * IMG_W);
    const float* iimg = inp + (size_t)b * (IMG_H * IMG_W);

    // ---- stage targets 36x36 (conv zero padding applied here) ----
    for (int idx = tid; idx < 36 * 36; idx += 256) {
        int a = idx / 36, c = idx - a * 36;
        int gy = y0 - 2 + a, gx = x0 - 2 + c;
        float v = 0.f;
        if ((unsigned)gy < IMG_H && (unsigned)gx < IMG_W)
            v = timg[gy * IMG_W + gx];
        st[a][c] = v;
    }
    // ---- stage inputs 34x34 ----
    for (int idx = tid; idx < 34 * 34; idx += 256) {
        int a = idx / 34, c = idx - a * 34;
        int gy = y0 - 1 + a, gx = x0 - 1 + c;
        float v = 0.f;
        if ((unsigned)gy < IMG_H && (unsigned)gx < IMG_W)
            v = iimg[gy * IMG_W + gx];
        si[a][c] = v;
    }
    __syncthreads();

    // ---- target edge magnitude on halo-1 region (34x34) ----
    // Out-of-image positions get -1 so the 3x3 max (maxpool with -inf padding
    // in the reference) ignores them; real edge values are >= 0.
    for (int idx = tid; idx < 34 * 34; idx += 256) {
        int a = idx / 34, c = idx - a * 34;
        int gy = y0 - 1 + a, gx = x0 - 1 + c;
        float v = -1.0f;
        if ((unsigned)gy < IMG_H && (unsigned)gx < IMG_W) {
            float px = (st[a    ][c + 2] - st[a    ][c])
                + 2.f * (st[a + 1][c + 2] - st[a + 1][c])
                +       (st[a + 2][c + 2] - st[a + 2][c]);
            float py = (st[a + 2][c] + 2.f * st[a + 2][c + 1] + st[a + 2][c + 2])
                     - (st[a    ][c] + 2.f * st[a    ][c + 1] + st[a    ][c + 2]);
            v = sqrtf(px * px + py * py);
        }
        te[a][c] = v;
    }
    __syncthreads();

    // ---- per-thread: 4 output pixels of diff^2 ----
    float acc = 0.f;
    for (int k = 0; k < 4; ++k) {
        int idx = k * 256 + tid;
        int ty = idx >> 5, tx = idx & 31;

        float px = (si[ty    ][tx + 2] - si[ty    ][tx])
            + 2.f * (si[ty + 1][tx + 2] - si[ty + 1][tx])
            +       (si[ty + 2][tx + 2] - si[ty + 2][tx]);
        float py = (si[ty + 2][tx] + 2.f * si[ty + 2][tx + 1] + si[ty + 2][tx + 2])
                 - (si[ty    ][tx] + 2.f * si[ty    ][tx + 1] + si[ty    ][tx + 2]);
        float pe = sqrtf(px * px + py * py);

        float m = te[ty][tx];
        m = fmaxf(m, te[ty    ][tx + 1]); m = fmaxf(m, te[ty    ][tx + 2]);
        m = fmaxf(m, te[ty + 1][tx    ]); m = fmaxf(m, te[ty + 1][tx + 1]);
        m = fmaxf(m, te[ty + 1][tx + 2]);
        m = fmaxf(m, te[ty + 2][tx    ]); m = fmaxf(m, te[ty + 2][tx + 1]);
        m = fmaxf(m, te[ty + 2][tx + 2]);

        float d = m * (pe - 1.0f);      // pe*tep - tep
        acc += d * d;
    }

    // ---- wave32 reduction on the matrix pipe: D = A(16x4) x ones(4x16) ----
    // A lane layout: lanes 0-15 carry rows M=lane at K={0,1}; lanes 16-31 at
    // K={2,3}. With B=1, C=0: D[m][n] = acc(m) + acc(m+16) for all n.
    // Full wave: no divergence here, EXEC is all ones as WMMA requires.
    v2f a2; a2.x = acc; a2.y = 0.f;
    v2f b2; b2.x = 1.f; b2.y = 1.f;
    v8f c8 = {};
    c8 = __builtin_amdgcn_wmma_f32_16x16x4_f32(
        /*neg_a=*/false, a2, /*neg_b=*/false, b2,
        /*c_mod=*/(short)0, c8, /*reuse_a=*/false, /*reuse_b=*/false);
    // lanes 0-15 hold M=0..7 in v[0..7]; lanes 16-31 hold M=8..15.
    float half = c8[0] + c8[1] + c8[2] + c8[3] + c8[4] + c8[5] + c8[6] + c8[7];
    float wtot = half + __shfl_xor(half, 16, 32);

    int lane = tid & 31, wave = tid >> 5;
    if (lane == 0) wsum[wave] = wtot;
    __syncthreads();
    if (tid == 0) {
        float s = 0.f;
        for (int i = 0; i < 8; ++i) s += wsum[i];
        partial[blk] = s;
    }
}

// Deterministic final reduction: fixed traversal order, single block.
__global__ __launch_bounds__(256)
void reduce_partials(const float* __restrict__ partial,
                     float* __restrict__ out, int n, float scale)
{
    __shared__ float s[256];
    int tid = threadIdx.x;
    float acc = 0.f;
    for (int i = tid; i < n; i += 256) acc += partial[i];
    s[tid] = acc;
    __syncthreads();
    if (tid == 0) {
        float tot = 0.f;
        for (int i = 0; i < 256; ++i) tot += s[i];
        out[0] = tot * scale;
    }
}

extern "C" void kernel_launch(void* const* d_in, const int* in_sizes, int n_in,
                              void* d_out, int out_size, void* d_ws, size_t ws_size,
                              hipStream_t stream)
{
    const float* inp = (const float*)d_in[0];
    const float* tgt = (const float*)d_in[1];
    float* partial   = (float*)d_ws;

    const int B = in_sizes[0] >> 20;            // elements / (1024*1024)
    const int nblocks = B * (IMG_H / TILE) * (IMG_W / TILE);   // 16384 for B=16

    edge_loss_tile<<<nblocks, 256, 0, stream>>>(inp, tgt, partial);

    const float scale = 1.0f / ((float)B * (float)IMG_H * (float)IMG_W);
    reduce_partials<<<1, 256, 0, stream>>>(partial, (float*)d_out, nblocks, scale);
}